// SNN_Baseline_90314572300872
// MI455X (gfx1250) — compile-verified
//
#include <hip/hip_runtime.h>

typedef __attribute__((ext_vector_type(2))) float v2f;
typedef __attribute__((ext_vector_type(8))) float v8f;

// Opaque-ify a 32-bit LDS element offset so instcombine cannot reassociate
// region-base constants into per-load offsets (keeps DS immediates < 64KB).
#define LAUNDER_U32(v) asm volatile("" : "+v"(v))

namespace {
constexpr int   kL      = 16;
constexpr float kBeta   = 0.9f;
constexpr float kThresh = 1.0f;
constexpr int   kB      = 65536;
constexpr int   kDin    = 64;
constexpr int   kH      = 128;

constexpr int kWaves        = 4;
constexpr int kThreads      = kWaves * 32;
constexpr int kRowsPerWave  = 16;
constexpr int kRowsPerBlock = kWaves * kRowsPerWave;   // 64 -> 1024 blocks

// Padded LDS row strides (floats) to avoid 64-bank conflicts.
constexpr int kWsynStride = 136;   // row pair (R,R+1)@half-wave -> disjoint banks
constexpr int kWinStride  = 136;
constexpr int kHStride    = 132;   // col read across 16 rows -> bank shift 4/row

// Flat LDS layout (element offsets).
constexpr unsigned kWinOff  = 0;                                   // 128x136
constexpr unsigned kWsynOff = kWinOff + (unsigned)(kH * kWinStride);     // 17408
constexpr unsigned kHOff    = kWsynOff + (unsigned)(kDin * kWsynStride); // 26112
constexpr unsigned kHWaveF  = (unsigned)(kRowsPerWave * kHStride);       // 2112
constexpr unsigned kTotalF  = kHOff + (unsigned)kWaves * kHWaveF;        // 34560
} // namespace

__global__ __launch_bounds__(kThreads)
void gif_snn_fused(const float* __restrict__ x,
                   const float* __restrict__ W_syn,
                   const float* __restrict__ b_syn,
                   const float* __restrict__ W_in,
                   const float* __restrict__ b_in,
                   float* __restrict__ out) {
  __shared__ __align__(16) float lds[kTotalF];                 // 138240 B

  const int tid = threadIdx.x;

  // ---- Cooperative weight staging into LDS (re-padded rows) ----
  for (int i = tid; i < kDin * kH; i += kThreads) {
    const int r = i >> 7, c = i & 127;
    lds[kWsynOff + r * kWsynStride + c] = W_syn[i];
  }
  for (int i = tid; i < kH * kH; i += kThreads) {
    const int r = i >> 7, c = i & 127;
    lds[kWinOff + r * kWinStride + c] = W_in[i];
  }
  __syncthreads();

  const int wave  = tid >> 5;
  const int lane  = tid & 31;
  const int m     = lane & 15;            // row (A) / col (B,C/D) within 16
  const int khalf = (lane >> 4) << 1;     // 0 or 2 : K sub-pair for A/B frags
  const int rhalf = (lane >> 4) << 3;     // 0 or 8 : row offset for C/D frags
  const int row_base = blockIdx.x * kRowsPerBlock + wave * kRowsPerWave;

  // =======================================================================
  // GEMM1: h[16x128] = x[16x64] @ W_syn[64x128] + b_syn
  // B reads: laundered base offset + compile-time consts (<= 33632 B -> imm)
  // =======================================================================
  v2f a1[16];
  {
    const float* xr = x + (size_t)(row_base + m) * kDin + khalf;
#pragma unroll
    for (int k = 0; k < 16; ++k)
      a1[k] = *(const v2f*)(xr + 4 * k);
  }

  unsigned wsb = kWsynOff + (unsigned)(khalf * kWsynStride + m);
  LAUNDER_U32(wsb);
  unsigned hso = kHOff + (unsigned)wave * kHWaveF
               + (unsigned)(rhalf * kHStride + m);
  LAUNDER_U32(hso);

#pragma unroll
  for (int nt = 0; nt < 8; ++nt) {
    v8f acc = {};
#pragma unroll
    for (int k = 0; k < 16; ++k) {
      v2f b;
      b.x = lds[wsb + (4 * k) * kWsynStride + nt * 16];
      b.y = lds[wsb + (4 * k + 1) * kWsynStride + nt * 16];
      acc = __builtin_amdgcn_wmma_f32_16x16x4_f32(
          false, a1[k], false, b, (short)0, acc, false, false);
    }
    const float bias = b_syn[nt * 16 + m];
    // Spill D fragment (row = r+rhalf, col = nt*16+m) to padded LDS
#pragma unroll
    for (int r = 0; r < 8; ++r)
      lds[hso + r * kHStride + nt * 16] = acc[r] + bias;
  }

  // =======================================================================
  // GEMM2: I[16x128] = h[16x128] @ W_in[128x128] + b_in
  // W_in spans 68KB (> DS imm range), so two fully-unrolled 16-kstep chunks
  // with the B base re-laundered per chunk; in-chunk consts <= 33632 B.
  // =======================================================================
  v8f acc2[8];
  {
    const v8f zero = {};
#pragma unroll
    for (int nt = 0; nt < 8; ++nt) acc2[nt] = zero;
  }
  unsigned hao = kHOff + (unsigned)wave * kHWaveF
               + (unsigned)(m * kHStride + khalf);
  LAUNDER_U32(hao);

  unsigned wib = kWinOff + (unsigned)(khalf * kWinStride + m);
#pragma unroll
  for (int kc = 0; kc < 2; ++kc) {
    LAUNDER_U32(wib);
    const unsigned wib_c = wib;
#pragma unroll
    for (int kk = 0; kk < 16; ++kk) {
      const int k = kc * 16 + kk;
      const v2f a = *(const v2f*)(&lds[hao + 4 * k]);   // ds_load_b64, imm
#pragma unroll
      for (int nt = 0; nt < 8; ++nt) {
        v2f b;
        b.x = lds[wib_c + (4 * kk) * kWinStride + nt * 16];
        b.y = lds[wib_c + (4 * kk + 1) * kWinStride + nt * 16];
        acc2[nt] = __builtin_amdgcn_wmma_f32_16x16x4_f32(
            false, a, false, b, (short)0, acc2[nt], false, false);
      }
    }
    wib += 64u * kWinStride;    // next 64 W_in rows (16 k-steps)
  }

#pragma unroll
  for (int nt = 0; nt < 8; ++nt) {
    const float b2 = b_in[nt * 16 + m];
#pragma unroll
    for (int r = 0; r < 8; ++r) acc2[nt][r] += b2;
  }

  // =======================================================================
  // GIF recurrence. s_{t-1} == (v_{t-1} >= THRESH) with v0=0, so the previous
  // spike is recomputed from the live membrane potential (no 64-reg s array).
  // One per-lane base pointer; all 64 stores/timestep use constant 24-bit
  // immediates (<= 4032 B); pointer bumps one plane per t. Spike stream is
  // write-once (570 MB) -> non-temporal stores keep it out of L2.
  // =======================================================================
  float vv[8][8];
#pragma unroll
  for (int nt = 0; nt < 8; ++nt)
#pragma unroll
    for (int r = 0; r < 8; ++r) vv[nt][r] = 0.0f;

  float* ot = out + (size_t)(row_base + rhalf) * kH + m;  // per-lane base

#pragma unroll 1
  for (int t = 0; t < kL; ++t) {
#pragma unroll
    for (int nt = 0; nt < 8; ++nt) {
#pragma unroll
      for (int r = 0; r < 8; ++r) {
        const float s_prev = (vv[nt][r] >= kThresh) ? 1.0f : 0.0f;
        const float v = kBeta * vv[nt][r] + acc2[nt][r] - kThresh * s_prev;
        const float s = (v >= kThresh) ? 1.0f : 0.0f;
        vv[nt][r] = v;
        __builtin_nontemporal_store(s, ot + r * kH + nt * 16);
      }
    }
    ot += (size_t)kB * kH;   // advance one timestep plane
  }

  // v_final plane sits after the L spike planes; same constant-imm pattern.
  float* ovf = out + ((size_t)kL * kB + (size_t)(row_base + rhalf)) * kH + m;
#pragma unroll
  for (int nt = 0; nt < 8; ++nt)
#pragma unroll
    for (int r = 0; r < 8; ++r)
      __builtin_nontemporal_store(vv[nt][r], ovf + r * kH + nt * 16);
}

extern "C" void kernel_launch(void* const* d_in, const int* in_sizes, int n_in,
                              void* d_out, int out_size, void* d_ws, size_t ws_size,
                              hipStream_t stream) {
  const float* x     = (const float*)d_in[0];
  const float* W_syn = (const float*)d_in[1];
  const float* b_syn = (const float*)d_in[2];
  const float* W_in  = (const float*)d_in[3];
  const float* b_in  = (const float*)d_in[4];
  float* out = (float*)d_out;

  dim3 grid(kB / kRowsPerBlock);   // 1024
  dim3 block(kThreads);            // 128 = 4 waves
  hipLaunchKernelGGL(gif_snn_fused, grid, block, 0, stream,
                     x, W_syn, b_syn, W_in, b_in, out);
}